// PartAttentionBlock_60954175865295
// MI455X (gfx1250) — compile-verified
//
#include <hip/hip_runtime.h>
#include <hip/hip_bf16.h>

typedef __attribute__((ext_vector_type(2))) float v2f;
typedef __attribute__((ext_vector_type(8))) float v8f;

#define XSTRIDE 268    // xn rows: 268 ≡ 12 (mod 64) -> 12*m is a permutation over m=0..15 (A-frag conflict-free)
#define ASTRIDE 68     // A_w rows: 68 ≡ 4  -> 4*m permutation (A-frag conflict-free)
#define YPSTRIDE 544   // y packed pair-rows: 544 ≡ 32, 2*544 ≡ 0 -> g0/g1 b64 reads hit disjoint 32-bank halves
#define EPSTRIDE 160   // eb packed pair-rows: 160 ≡ 32, 2*160 ≡ 0 -> same property
#define SXFLOATS 17408 // max(64*XSTRIDE=17152, 32*YPSTRIDE=17408)

// ---------------------------------------------------------------------------
// Kernel 1: pack Wp^T into K-pair-interleaved B layout:
//   WpP[p*512 + o*2 + j] = Wp[o*256 + 2p + j]   (p=0..127, o=0..255, j=0..1)
// A B-fragment for (k,g,col) is then the 8B word at WpP + (2k+g)*512 + 2*col.
// ---------------------------------------------------------------------------
__global__ void k_pack_wp(const float* __restrict__ Wp, float* __restrict__ WpP) {
    int i = blockIdx.x * 256 + threadIdx.x;      // 32768 total
    int p = i >> 8, o = i & 255;
    v2f v = *(const v2f*)(Wp + o * 256 + 2 * p); // 8B-aligned pair read
    *(v2f*)(WpP + p * 512 + o * 2) = v;
}

// ---------------------------------------------------------------------------
// Kernel 2: A[w] = deg[w] @ eb[w]  (256 windows, 64x64 fp32).
// deg staged row-major (A operand); eb staged K-pair packed (B operand).
// ---------------------------------------------------------------------------
__global__ __launch_bounds__(128) void k_compute_A(const float* __restrict__ deg,
                                                   const float* __restrict__ eb,
                                                   float* __restrict__ A) {
    __shared__ float s_deg[64 * ASTRIDE];
    __shared__ float s_ebP[32 * EPSTRIDE];
    const int w = blockIdx.x;
    const int t = threadIdx.x;                   // 0..127
    const float* degw = deg + (size_t)w * 4096;
    const float* ebw  = eb  + (size_t)w * 4096;

    {
        int r = t >> 1, c0 = (t & 1) * 32;
        const float4* sd = (const float4*)(degw + r * 64 + c0);
        float4* dd = (float4*)(s_deg + r * ASTRIDE + c0);
#pragma unroll
        for (int j = 0; j < 8; ++j) dd[j] = sd[j];
        // eb -> packed: pair p = r>>1, j = r&1
        float* ep = s_ebP + (r >> 1) * EPSTRIDE + (r & 1);
        const float* es = ebw + r * 64 + c0;
#pragma unroll
        for (int c = 0; c < 32; ++c) ep[(c0 + c) * 2] = es[c];
    }
    __syncthreads();

    const int lane = t & 31, wid = t >> 5;       // wid == mt
    const int m = lane & 15, g = lane >> 4;
    float* outw = A + (size_t)w * 4096;
    const float* ap = s_deg + (wid * 16 + m) * ASTRIDE;

    for (int nt = 0; nt < 4; ++nt) {
        v8f acc = {};
        for (int k = 0; k < 16; ++k) {
            v2f a = *(const v2f*)(ap + 4 * k + 2 * g);
            v2f b = *(const v2f*)(s_ebP + (2 * k + g) * EPSTRIDE + (nt * 16 + m) * 2);
            acc = __builtin_amdgcn_wmma_f32_16x16x4_f32(false, a, false, b,
                                                        (short)0, acc, false, false);
        }
#pragma unroll
        for (int r = 0; r < 8; ++r)
            outw[(wid * 16 + r + 8 * g) * 64 + nt * 16 + m] = acc[r];
    }
}

// ---------------------------------------------------------------------------
// Kernel 3: per-window fused LN + (xn @ WpT + bp) + (A_w @ y), window scatter.
// One block (8 waves) per window n = b*256 + fi*4 + wi.
// Wave: mt = wid&3, column tiles nt in [(wid>>2)*8, +8).
// LDS: s_x (xn 64x268, reused as packed y 32x544) + s_A (64x68) = 87040 B.
// ---------------------------------------------------------------------------
__global__ __launch_bounds__(256) void k_main(const float* __restrict__ x,
                                              const float* __restrict__ gamma1,
                                              const float* __restrict__ beta1,
                                              const float* __restrict__ WpP,
                                              const float* __restrict__ bp,
                                              const float* __restrict__ A,
                                              float* __restrict__ out) {
    extern __shared__ float smem[];
    float* s_x = smem;                       // 64*XSTRIDE, later 32*YPSTRIDE (packed y)
    float* s_A = smem + SXFLOATS;            // 64*ASTRIDE

    const int n   = blockIdx.x;
    const int b   = n >> 8, rem = n & 255;
    const int fi  = rem >> 2, wi = rem & 3;
    const int t   = threadIdx.x, lane = t & 31, wid = t >> 5;

    // ---- stage window xw and A_w into LDS ----
    {
        int l = t >> 2, c0 = (t & 3) * 64;   // row l, 64-float segment
        int tp = l >> 4, ws = l & 15;
        const float4* src = (const float4*)(x +
            (((size_t)(b * 256 + fi * 4 + tp)) * 64 + wi * 16 + ws) * 256 + c0);
        float4* dst = (float4*)(s_x + l * XSTRIDE + c0);
#pragma unroll
        for (int j = 0; j < 16; ++j) dst[j] = src[j];

        const float* Aw = A + (size_t)rem * 4096;
        int ar = t >> 2, ac = (t & 3) * 16;
        const float4* as = (const float4*)(Aw + ar * 64 + ac);
        float4* ad = (float4*)(s_A + ar * ASTRIDE + ac);
#pragma unroll
        for (int j = 0; j < 4; ++j) ad[j] = as[j];
    }
    __syncthreads();

    // ---- LayerNorm (biased var, eps=1e-5), wave wid handles rows wid*8..+8 ----
    for (int rr = 0; rr < 8; ++rr) {
        float* rp = s_x + (wid * 8 + rr) * XSTRIDE;
        float s = 0.f, ss = 0.f, vals[8];
#pragma unroll
        for (int i = 0; i < 8; ++i) {
            float v = rp[lane + 32 * i];
            vals[i] = v; s += v; ss += v * v;
        }
#pragma unroll
        for (int msk = 16; msk >= 1; msk >>= 1) {
            s  += __shfl_xor(s,  msk, 32);
            ss += __shfl_xor(ss, msk, 32);
        }
        float mu  = s * (1.f / 256.f);
        float var = ss * (1.f / 256.f) - mu * mu;
        float inv = rsqrtf(var + 1e-5f);
#pragma unroll
        for (int i = 0; i < 8; ++i) {
            int c = lane + 32 * i;
            rp[c] = (vals[i] - mu) * inv * gamma1[c] + beta1[c];
        }
    }
    __syncthreads();

    // ---- GEMM1: y = xn @ WpT  (64x256 @ 256x256), tiles (mt, ntb..ntb+8) ----
    const int mt = wid & 3, ntb = (wid >> 2) * 8;
    const int m = lane & 15, g = lane >> 4;
    v8f acc[8] = {};
    const float* ax = s_x + (mt * 16 + m) * XSTRIDE;
    for (int k = 0; k < 64; ++k) {
        v2f a = *(const v2f*)(ax + 4 * k + 2 * g);              // ds_load_b64
        const float* wb = WpP + (size_t)(2 * k + g) * 512;
#pragma unroll
        for (int nti = 0; nti < 8; ++nti) {
            v2f bf = *(const v2f*)(wb + ((ntb + nti) * 16 + m) * 2);  // global_load_b64
            acc[nti] = __builtin_amdgcn_wmma_f32_16x16x4_f32(false, a, false, bf,
                                                             (short)0, acc[nti], false, false);
        }
    }
    __syncthreads();                          // everyone done reading xn

    // ---- write y (+bp) into s_x as K-pair packed B layout ----
#pragma unroll
    for (int nti = 0; nti < 8; ++nti) {
        int col = (ntb + nti) * 16 + m;
        float bpv = bp[col];
#pragma unroll
        for (int r = 0; r < 8; ++r) {
            int row = mt * 16 + r + 8 * g;
            s_x[(row >> 1) * YPSTRIDE + col * 2 + (row & 1)] = acc[nti][r] + bpv;
        }
    }
    __syncthreads();

    // ---- GEMM2: z = A_w @ y  (64x64 @ 64x256) ----
    v8f zro = {};
#pragma unroll
    for (int i = 0; i < 8; ++i) acc[i] = zro;
    const float* aA = s_A + (mt * 16 + m) * ASTRIDE;
    for (int k = 0; k < 16; ++k) {
        v2f a = *(const v2f*)(aA + 4 * k + 2 * g);              // ds_load_b64
        const float* yb = s_x + (2 * k + g) * YPSTRIDE;
#pragma unroll
        for (int nti = 0; nti < 8; ++nti) {
            v2f bf = *(const v2f*)(yb + ((ntb + nti) * 16 + m) * 2);  // ds_load_b64, conflict-free
            acc[nti] = __builtin_amdgcn_wmma_f32_16x16x4_f32(false, a, false, bf,
                                                             (short)0, acc[nti], false, false);
        }
    }

    // ---- scatter z with window_reverse ----
#pragma unroll
    for (int r = 0; r < 8; ++r) {
        int l = mt * 16 + r + 8 * g;
        int tp = l >> 4, ws = l & 15;
        float* dst = out + (((size_t)(b * 256 + fi * 4 + tp)) * 64 + wi * 16 + ws) * 256;
#pragma unroll
        for (int nti = 0; nti < 8; ++nti)
            dst[(ntb + nti) * 16 + m] = acc[nti][r];
    }
}

// ---------------------------------------------------------------------------
extern "C" void kernel_launch(void* const* d_in, const int* in_sizes, int n_in,
                              void* d_out, int out_size, void* d_ws, size_t ws_size,
                              hipStream_t stream) {
    const float* x      = (const float*)d_in[0];
    const float* gamma1 = (const float*)d_in[1];
    const float* beta1  = (const float*)d_in[2];
    const float* Wp     = (const float*)d_in[3];
    const float* bp     = (const float*)d_in[4];
    const float* eb     = (const float*)d_in[5];
    const float* deg    = (const float*)d_in[6];
    float* out = (float*)d_out;

    float* WpP  = (float*)d_ws;              // 128*512 floats (packed Wp^T)
    float* Abuf = WpP + 128 * 512;           // 256*64*64 floats

    size_t smem = (size_t)(SXFLOATS + 64 * ASTRIDE) * sizeof(float);   // 87040 B
    (void)hipFuncSetAttribute((const void*)k_main,
                              hipFuncAttributeMaxDynamicSharedMemorySize, (int)smem);

    k_pack_wp  <<<128, 256, 0, stream>>>(Wp, WpP);
    k_compute_A<<<256, 128, 0, stream>>>(deg, eb, Abuf);
    k_main     <<<2048, 256, smem, stream>>>(x, gamma1, beta1, WpP, bp, Abuf, out);
}